// ChannelSelfAttn_22711787061804
// MI455X (gfx1250) — compile-verified
//
#include <hip/hip_runtime.h>
#include <math.h>

typedef __attribute__((ext_vector_type(2))) float v2f;
typedef __attribute__((ext_vector_type(8))) float v8f;

#define BATCH 32
#define C 64
#define T 4000
#define F 16

// ---------------------------------------------------------------------------
// Stage 1: per-batch Gram matrix G[b] = X_b * X_b^T  (64x64, K=4000) + row sums
// One wave (32 threads) per 16x16 output tile; v_wmma_f32_16x16x4_f32.
// grid = (4 i-tiles, 4 j-tiles, 32 batches), block = 32.
// ---------------------------------------------------------------------------
__global__ void gram_rowsum_kernel(const float* __restrict__ x,
                                   float* __restrict__ G,
                                   float* __restrict__ S) {
    const int b  = blockIdx.z;
    const int i0 = blockIdx.x * 16;
    const int j0 = blockIdx.y * 16;
    const int lane = threadIdx.x;        // 0..31 (wave32)
    const int lm   = lane & 15;          // 0..15
    const int h    = lane >> 4;          // 0 or 1

    const float* xb   = x + (size_t)b * C * T;
    // A[m][kk] = X[i0+m][k+kk]; lane(h,lm) holds kk = 2h, 2h+1 of row i0+lm
    const float* arow = xb + (size_t)(i0 + lm) * T + 2 * h;
    // B[kk][n] = X[j0+n][k+kk]; same per-lane pattern on row j0+lm
    const float* brow = xb + (size_t)(j0 + lm) * T + 2 * h;

    v8f acc = {};
    float rsum = 0.0f;

    #pragma unroll 4
    for (int k = 0; k < T; k += 4) {
        v2f av = *(const v2f*)(arow + k);
        v2f bv = *(const v2f*)(brow + k);
        rsum += av.x + av.y;
        acc = __builtin_amdgcn_wmma_f32_16x16x4_f32(
            /*neg_a=*/false, av, /*neg_b=*/false, bv,
            /*c_mod=*/(short)0, acc, /*reuse_a=*/false, /*reuse_b=*/false);
    }

    // D layout: vgpr v -> row M = v + 8*h, col N = lm
    float* Gt = G + ((size_t)b * C + i0) * C + j0;
    #pragma unroll
    for (int v = 0; v < 8; ++v) {
        Gt[(size_t)(v + 8 * h) * C + lm] = acc[v];
    }

    // Row sums: lane(h,lm) summed columns {2h,2h+1} mod 4 of row i0+lm.
    // Combine halves; only the j0==0 blocks own the store (each row once).
    float other = __shfl_down(rsum, 16, 32);
    if (blockIdx.y == 0 && h == 0) {
        S[b * C + i0 + lm] = rsum + other;
    }
}

// ---------------------------------------------------------------------------
// Stage 2: e = a*G + c1*S_i + c2*S_j + c0*T; min-max normalize per row;
// softmax per row; overwrite G with P. grid = 32 batches, block = 64 rows.
// ---------------------------------------------------------------------------
__global__ void normalize_softmax_kernel(const float* __restrict__ w1,
                                         const float* __restrict__ b1,
                                         const float* __restrict__ w2,
                                         const float* __restrict__ b2,
                                         float* __restrict__ G,   // in: Gram, out: P
                                         const float* __restrict__ S) {
    const int b = blockIdx.x;
    const int i = threadIdx.x;           // 0..63

    float ca = 0.0f, c1 = 0.0f, c2 = 0.0f, c0 = 0.0f;
    #pragma unroll
    for (int f = 0; f < F; ++f) {
        ca += w1[f] * w2[f];
        c1 += w1[f] * b2[f];
        c2 += b1[f] * w2[f];
        c0 += b1[f] * b2[f];
    }

    float* Gb = G + (size_t)b * C * C;
    const float* Sb = S + b * C;
    const float Si = Sb[i];
    const float cterm = c1 * Si + c0 * (float)T;

    float e[C];
    float emax = -INFINITY, emin = INFINITY;
    for (int j = 0; j < C; ++j) {
        float val = ca * Gb[(size_t)i * C + j] + cterm + c2 * Sb[j];
        e[j] = val;
        emax = fmaxf(emax, val);
        emin = fminf(emin, val);
    }
    const float inv = 1.0f / (emax - emin + 1e-8f);

    float nmax = -INFINITY;
    for (int j = 0; j < C; ++j) {
        e[j] = (e[j] - emin) * inv;
        nmax = fmaxf(nmax, e[j]);
    }
    float sum = 0.0f;
    for (int j = 0; j < C; ++j) {
        float p = expf(e[j] - nmax);
        e[j] = p;
        sum += p;
    }
    const float invsum = 1.0f / sum;
    for (int j = 0; j < C; ++j) {
        Gb[(size_t)i * C + j] = e[j] * invsum;   // in place: own row only
    }
}

// ---------------------------------------------------------------------------
// Stage 3: out[b,i,t] = gamma * (P[b] @ X_b)[i,t] + x[b,i,t]
// One wave per 16x16 (i,t) tile, K = 64 in steps of 4 via wmma f32 16x16x4.
// grid = (4 i-tiles, 250 t-tiles, 32 batches), block = 32.
// ---------------------------------------------------------------------------
__global__ void attn_apply_kernel(const float* __restrict__ x,
                                  const float* __restrict__ P,
                                  const float* __restrict__ gamma,
                                  float* __restrict__ out) {
    const int b  = blockIdx.z;
    const int i0 = blockIdx.x * 16;
    const int t0 = blockIdx.y * 16;
    const int lane = threadIdx.x;
    const int lm   = lane & 15;
    const int h    = lane >> 4;

    const float* xb = x + (size_t)b * C * T;
    const float* Pl = P + (size_t)b * C * C + (size_t)(i0 + lm) * C + 2 * h;

    v8f acc = {};
    #pragma unroll
    for (int k = 0; k < C; k += 4) {
        // A[m][kk] = P[i0+m][k+kk]
        v2f av = *(const v2f*)(Pl + k);
        // B[kk][n] = X[k+kk][t0+n]; lanes 0-15 / 16-31 each hit one 64B line
        v2f bv;
        bv.x = xb[(size_t)(k + 2 * h)     * T + t0 + lm];
        bv.y = xb[(size_t)(k + 2 * h + 1) * T + t0 + lm];
        acc = __builtin_amdgcn_wmma_f32_16x16x4_f32(
            false, av, false, bv, (short)0, acc, false, false);
    }

    const float g = gamma[0];
    float* ob = out + (size_t)b * C * T;
    #pragma unroll
    for (int v = 0; v < 8; ++v) {
        const size_t idx = (size_t)(i0 + v + 8 * h) * T + (t0 + lm);
        ob[idx] = g * acc[v] + xb[idx];
    }
}

// ---------------------------------------------------------------------------
extern "C" void kernel_launch(void* const* d_in, const int* in_sizes, int n_in,
                              void* d_out, int out_size, void* d_ws, size_t ws_size,
                              hipStream_t stream) {
    const float* x     = (const float*)d_in[0];
    const float* w1    = (const float*)d_in[1];
    const float* b1    = (const float*)d_in[2];
    const float* w2    = (const float*)d_in[3];
    const float* b2    = (const float*)d_in[4];
    const float* gamma = (const float*)d_in[5];
    float* out = (float*)d_out;

    float* G = (float*)d_ws;                 // 32*64*64 floats = 512 KB (Gram -> P)
    float* S = G + (size_t)BATCH * C * C;    // 32*64 floats    =   8 KB (row sums)

    dim3 g1(4, 4, BATCH);
    gram_rowsum_kernel<<<g1, 32, 0, stream>>>(x, G, S);

    normalize_softmax_kernel<<<BATCH, C, 0, stream>>>(w1, b1, w2, b2, G, S);

    dim3 g3(4, T / 16, BATCH);
    attn_apply_kernel<<<g3, 32, 0, stream>>>(x, G, gamma, out);
}